// BaseSparseConn_47571057770801
// MI455X (gfx1250) — compile-verified
//
#include <hip/hip_runtime.h>

// ---------------------------------------------------------------------------
// COO SpMM: out[b,d] = sum_e values[e] * x[b, col[e]] (row[e]==d) + bias[d]
// Memory-bound (AI ~1.3 flop/byte, ~77MB total < 192MB L2). Strategy:
//   K1: zero outT (DST x 16) accumulator in workspace
//   K2: xT (SRC x 16) = x^T  -> each src column is 64B contiguous
//   K3: one thread / edge: NT-stream vals+indices, gather 64B of xT,
//       16x global_atomic_add_f32 into outT (L2-resident)
//   K4: out = outT^T + bias per 16x16 tile via V_WMMA_F32_16X16X4_F32:
//       D = sum_q I_q(16x4) * B_q(4x16) + C(bias broadcast), where B_q is
//       loaded straight from outT in B-operand layout -> the A-layout vs
//       D-layout striping performs the transpose; bias is the C accumulator.
// ---------------------------------------------------------------------------

typedef float v2f __attribute__((ext_vector_type(2)));
typedef float v8f __attribute__((ext_vector_type(8)));

#define BATCH 16

__global__ void zero_f32(float* __restrict__ p, int n) {
    int i = blockIdx.x * blockDim.x + threadIdx.x;
    if (i < n) p[i] = 0.0f;
}

// xT[s*16 + b] = x[b*nsrc + s]; writes coalesced, reads L2-resident.
__global__ void transpose_x(const float* __restrict__ x,
                            float* __restrict__ xT, int nsrc) {
    int i = blockIdx.x * blockDim.x + threadIdx.x;
    if (i < nsrc * BATCH) {
        int s = i >> 4;
        int b = i & 15;
        xT[i] = x[b * nsrc + s];
    }
}

// One thread per edge. values/indices streamed once (non-temporal);
// xT gather is 4x b128 from L2; 16 f32 atomics (no-return) into outT.
__global__ void scatter_edges(const float* __restrict__ vals,
                              const int* __restrict__ idx,
                              const float* __restrict__ xT,
                              float* __restrict__ outT, int nnz) {
    int e = blockIdx.x * blockDim.x + threadIdx.x;
    if (e >= nnz) return;
    int r = __builtin_nontemporal_load(idx + e);        // dst row
    int c = __builtin_nontemporal_load(idx + nnz + e);  // src col
    float v = __builtin_nontemporal_load(vals + e);

    const float4* xs = (const float4*)(xT + (size_t)c * BATCH);
    float4 x0 = xs[0];
    float4 x1 = xs[1];
    float4 x2 = xs[2];
    float4 x3 = xs[3];

    float* o = outT + (size_t)r * BATCH;
    atomicAdd(o + 0,  v * x0.x); atomicAdd(o + 1,  v * x0.y);
    atomicAdd(o + 2,  v * x0.z); atomicAdd(o + 3,  v * x0.w);
    atomicAdd(o + 4,  v * x1.x); atomicAdd(o + 5,  v * x1.y);
    atomicAdd(o + 6,  v * x1.z); atomicAdd(o + 7,  v * x1.w);
    atomicAdd(o + 8,  v * x2.x); atomicAdd(o + 9,  v * x2.y);
    atomicAdd(o + 10, v * x2.z); atomicAdd(o + 11, v * x2.w);
    atomicAdd(o + 12, v * x3.x); atomicAdd(o + 13, v * x3.y);
    atomicAdd(o + 14, v * x3.z); atomicAdd(o + 15, v * x3.w);
}

// One wave per 16x16 tile of outT. D[b][d] = outT[d0+d][b] + bias[d0+d].
// A_q = identity slice delta(i, 4q+k) built in registers (A layout:
//   lane m=L%16 rows, VGPR pair holds K = 2*(L/16) + {0,1}).
// B_q[k][d] = outT[d0+d][4q+k] loaded per-lane as one 64-bit load
//   (B layout: lane d=L%16 columns, VGPR pair holds K = 2*(L/16) + {0,1}).
// C = bias[d0 + N] broadcast across all 8 accumulator VGPRs.
// D layout: VGPR r, lanes 0-15 -> (M=r, N=lane); lanes 16-31 -> (M=r+8).
__global__ void wmma_transpose_bias(const float* __restrict__ outT,
                                    const float* __restrict__ bias,
                                    float* __restrict__ out, int ndst) {
    int lane = threadIdx.x & 31;
    int wave = threadIdx.x >> 5;
    int tile = blockIdx.x * (blockDim.x >> 5) + wave;
    int d0 = tile * 16;
    if (d0 >= ndst) return;  // wave-uniform: EXEC stays all-ones for WMMA

    int m = lane & 15;   // row (A) / column (B,C,D) index within tile
    int h = lane >> 4;   // half-wave select
    int k0 = 2 * h;

    float bb = bias[d0 + m];
    v8f c = {bb, bb, bb, bb, bb, bb, bb, bb};

#pragma unroll
    for (int q = 0; q < 4; ++q) {
        v2f a, b;
        a.x = (m == 4 * q + k0)     ? 1.0f : 0.0f;
        a.y = (m == 4 * q + k0 + 1) ? 1.0f : 0.0f;
        // outT row (d0+m), elements [4q+k0, 4q+k0+1]; 8-byte aligned.
        b = *(const v2f*)(outT + (size_t)(d0 + m) * BATCH + 4 * q + k0);
        c = __builtin_amdgcn_wmma_f32_16x16x4_f32(
                /*neg_a=*/false, a, /*neg_b=*/false, b,
                /*c_mod=*/(short)0, c, /*reuse_a=*/false, /*reuse_b=*/false);
    }

#pragma unroll
    for (int r = 0; r < 8; ++r) {
        int brow = r + 8 * h;  // batch index
        out[(size_t)brow * ndst + d0 + m] = c[r];
    }
}

extern "C" void kernel_launch(void* const* d_in, const int* in_sizes, int n_in,
                              void* d_out, int out_size, void* d_ws, size_t ws_size,
                              hipStream_t stream) {
    const float* x    = (const float*)d_in[0];  // (B, SRC) f32
    const float* vals = (const float*)d_in[1];  // (NNZ,)  f32
    const float* bias = (const float*)d_in[2];  // (DST,)  f32
    const int*   idx  = (const int*)d_in[3];    // (2, NNZ) int32: [row; col]

    int nsrc = in_sizes[0] / BATCH;
    int nnz  = in_sizes[1];
    int ndst = in_sizes[2];

    float* out  = (float*)d_out;
    float* xT   = (float*)d_ws;                     // SRC x 16
    float* outT = xT + (size_t)nsrc * BATCH;        // DST x 16

    int nOutT = ndst * BATCH;
    zero_f32<<<(nOutT + 255) / 256, 256, 0, stream>>>(outT, nOutT);

    int nXT = nsrc * BATCH;
    transpose_x<<<(nXT + 255) / 256, 256, 0, stream>>>(x, xT, nsrc);

    scatter_edges<<<(nnz + 255) / 256, 256, 0, stream>>>(vals, idx, xT, outT, nnz);

    int tiles = (ndst + 15) / 16;  // 6250 tiles, 8 waves/block
    wmma_transpose_bias<<<(tiles + 7) / 8, 256, 0, stream>>>(outT, bias, out, ndst);
}